// FourierFeature_62586263437915
// MI455X (gfx1250) — compile-verified
//
#include <hip/hip_runtime.h>
#include <hip/hip_bf16.h>
#include <math.h>

typedef float v2f __attribute__((ext_vector_type(2)));
typedef float v8f __attribute__((ext_vector_type(8)));

// Problem constants
#define BB 16
#define CC 64
#define HH 256
#define WW 256
#define OD 125           // (256-8)/2+1
#define KTOT 4096        // 64*8*8
#define ND  (BB*OD*OD)   // 250000 distinct norm values
#define NFULL 4000000.0  // 256*125*125

// ws layout:
//  WA     : [16][1024][16][4] f32   offset 0            size 4,194,304 B
//  norm_d : [250000]          f32   offset 4,194,304    size 1,000,000 B
//  nf     : [250000]          f32   offset 5,194,304    size 1,000,000 B
//  S[2]   : doubles                 offset 6,194,304    size 16 B
#define WS_WA     0
#define WS_NORM   (4194304/4)
#define WS_NF     (5194304/4)
#define WS_S_BYTE 6194304

// ---------------------------------------------------------------------------
// Kernel 1: filter synthesis (rfft2 -> scale re/im -> irfft2, ortho) + write
// weights in WMMA-A layout WA[b][k>>2][m][k&3], k = c*64 + p*8 + q.
// One thread per (filter f, channel c); 16384 threads total.
// ---------------------------------------------------------------------------
__global__ __launch_bounds__(64)
void wgt_kernel(const float* __restrict__ z, const float* __restrict__ fc,
                float* __restrict__ WA, double* __restrict__ S) {
  int idx = blockIdx.x * 64 + threadIdx.x;
  if (idx == 0) { S[0] = 0.0; S[1] = 0.0; }
  if (idx >= BB * 16 * CC) return;
  int c = idx & 63;
  int f = idx >> 6;               // 0..255
  int b  = f >> 4;
  int n1 = (f >> 2) & 3;
  int n2 = f & 3;
  int m  = f & 15;

  float w[8][8];
  const float* zp = z + ((size_t)(b * 64 + c) * 32 + n1 * 8) * 32 + n2 * 8;
  #pragma unroll
  for (int p = 0; p < 8; ++p)
    #pragma unroll
    for (int q = 0; q < 8; ++q)
      w[p][q] = zp[p * 32 + q];

  const float ct[8] = {1.f, 0.70710678118654752f, 0.f, -0.70710678118654752f,
                       -1.f, -0.70710678118654752f, 0.f, 0.70710678118654752f};
  const float st[8] = {0.f, 0.70710678118654752f, 1.f, 0.70710678118654752f,
                       0.f, -0.70710678118654752f, -1.f, -0.70710678118654752f};

  // Forward rfft2 (ortho, total scale 1/8), keep v = 0..4, then apply coeffs.
  float Gre[8][5], Gim[8][5];
  for (int u = 0; u < 8; ++u)
    for (int v = 0; v < 5; ++v) {
      float re = 0.f, im = 0.f;
      for (int p = 0; p < 8; ++p)
        for (int q = 0; q < 8; ++q) {
          int t = (u * p + v * q) & 7;
          re += w[p][q] * ct[t];
          im -= w[p][q] * st[t];
        }
      re *= 0.125f; im *= 0.125f;
      Gre[u][v] = re * fc[(u * 5 + v) * 2 + 0];
      Gim[u][v] = im * fc[(u * 5 + v) * 2 + 1];
    }

  // Inverse step 1: full ifft along u (ortho, 1/sqrt(8)), e^{+2pi i u t/8}.
  const float is8 = 0.35355339059327373f;
  float Bre[8][5], Bim[8][5];
  for (int t2 = 0; t2 < 8; ++t2)
    for (int v = 0; v < 5; ++v) {
      float re = 0.f, im = 0.f;
      for (int u = 0; u < 8; ++u) {
        int t = (u * t2) & 7;
        re += Gre[u][v] * ct[t] - Gim[u][v] * st[t];
        im += Gre[u][v] * st[t] + Gim[u][v] * ct[t];
      }
      Bre[t2][v] = re * is8;
      Bim[t2][v] = im * is8;
    }

  // Inverse step 2: irfft along v (ortho); imag of v=0 and v=4 bins dropped.
  float* wb = WA + (size_t)b * 65536;
  for (int p = 0; p < 8; ++p)
    for (int q = 0; q < 8; ++q) {
      float sgn = (q & 1) ? -1.f : 1.f;
      float s = Bre[p][0] + sgn * Bre[p][4];
      for (int v = 1; v < 4; ++v) {
        int t = (v * q) & 7;
        s += 2.f * (Bre[p][v] * ct[t] - Bim[p][v] * st[t]);
      }
      s *= is8;
      int k = c * 64 + p * 8 + q;
      wb[(k >> 2) * 64 + m * 4 + (k & 3)] = s;
    }
}

// ---------------------------------------------------------------------------
// Kernel 2: norm_d[b][i][j] = sum_{c,kh,kw} x^2 over stride-2 8x8 patch,
// plus global double-precision sum / sum-of-squares via one atomic per block.
// One block (256 threads) per (b, output row i).
// ---------------------------------------------------------------------------
__global__ __launch_bounds__(256)
void norm_kernel(const float* __restrict__ x, float* __restrict__ norm_d,
                 double* __restrict__ S) {
  __shared__ float colsq[256];
  __shared__ double r1[256];
  __shared__ double r2[256];
  int t = threadIdx.x;
  int i = blockIdx.x;             // 0..124
  int b = blockIdx.y;             // 0..15

  const float* xg = x + (size_t)b * CC * HH * WW + (size_t)(2 * i) * WW;
  float s = 0.f;
  for (int c = 0; c < CC; ++c) {
    const float* xr = xg + (size_t)c * HH * WW;
    #pragma unroll
    for (int kh = 0; kh < 8; ++kh) {
      float v = xr[kh * WW + t];
      s += v * v;
    }
  }
  colsq[t] = s;
  __syncthreads();

  double v1 = 0.0, v2 = 0.0;
  if (t < OD) {
    float nrm = 0.f;
    #pragma unroll
    for (int kw = 0; kw < 8; ++kw) nrm += colsq[2 * t + kw];
    norm_d[((size_t)b * OD + i) * OD + t] = nrm;
    v1 = (double)nrm;
    v2 = (double)nrm * (double)nrm;
  }
  r1[t] = v1; r2[t] = v2;
  __syncthreads();
  for (int off = 128; off > 0; off >>= 1) {
    if (t < off) { r1[t] += r1[t + off]; r2[t] += r2[t + off]; }
    __syncthreads();
  }
  if (t == 0) {
    atomicAdd(&S[0], r1[0]);
    atomicAdd(&S[1], r2[0]);
  }
}

// ---------------------------------------------------------------------------
// Kernel 3: nf = 0.01 / sqrt(norm + std(norm_full, ddof=1)/10 + 1e-9).
// Replication across 16 out-channels multiplies SS by 16, mean unchanged.
// ---------------------------------------------------------------------------
__global__ __launch_bounds__(256)
void scale_kernel(const float* __restrict__ norm_d, const double* __restrict__ S,
                  float* __restrict__ nf) {
  int idx = blockIdx.x * 256 + threadIdx.x;
  double S1 = S[0], S2 = S[1];
  double SS = 16.0 * (S2 - S1 * S1 / (double)ND);
  double stdv = sqrt(SS / (NFULL - 1.0));
  float add = (float)(stdv * 0.1 + 1e-9);
  if (idx < ND) nf[idx] = 0.01f / sqrtf(norm_d[idx] + add);
}

// ---------------------------------------------------------------------------
// Kernel 4: grouped conv via V_WMMA_F32_16X16X4_F32.
// Block = 128 threads (4 waves). Wave tile: 16 out-channels x 32 out-columns
// (two accumulators sharing each A fragment -> 1.5 loads per WMMA).
// grid = (1, 125, 16): jtile = wave (0..3), 4x32 columns cover 125.
// ---------------------------------------------------------------------------
__global__ __launch_bounds__(128)
void conv_kernel(const float* __restrict__ x, const float* __restrict__ WA,
                 const float* __restrict__ nf, float* __restrict__ out) {
  const int lane = threadIdx.x & 31;
  const int wave = threadIdx.x >> 5;        // jtile 0..3
  const int i    = blockIdx.y;              // 0..124
  const int b    = blockIdx.z;              // 0..15
  const int j0   = wave * 32;
  const int n    = lane & 15;
  const int hi   = lane >> 4;               // 0: K+0/1 rows, 1: K+2/3 rows
  int ja = j0 + n;        if (ja > OD - 1) ja = OD - 1;   // clamp loads
  int jb = j0 + 16 + n;   if (jb > OD - 1) jb = OD - 1;

  const float* xg0 = x + (size_t)b * CC * HH * WW + (size_t)(2 * i) * WW + 2 * ja;
  const float* xg1 = x + (size_t)b * CC * HH * WW + (size_t)(2 * i) * WW + 2 * jb;
  const float* wg  = WA + (size_t)b * 65536;

  v8f acc0 = {};
  v8f acc1 = {};
  for (int c = 0; c < CC; ++c) {
    const float* xc0 = xg0 + (size_t)c * HH * WW;
    const float* xc1 = xg1 + (size_t)c * HH * WW;
    const float* wc  = wg + c * 1024;       // kc base = c*16, *64 floats
    #pragma unroll
    for (int kh = 0; kh < 8; ++kh) {
      #pragma unroll
      for (int kwg = 0; kwg < 2; ++kwg) {
        // A fragment: A[m][k0..k0+3]; lane<16 -> (k0,k0+1), lane>=16 -> (k0+2,k0+3)
        v2f a  = *(const v2f*)(wc + (kh * 2 + kwg) * 64 + n * 4 + hi * 2);
        // B fragments: B[k][n] = x[.., 2i+kh, 2j + kw]; kw = kwg*4 + hi*2 (+1)
        v2f b0 = *(const v2f*)(xc0 + kh * WW + kwg * 4 + hi * 2);
        v2f b1 = *(const v2f*)(xc1 + kh * WW + kwg * 4 + hi * 2);
        acc0 = __builtin_amdgcn_wmma_f32_16x16x4_f32(
            false, a, false, b0, (short)0, acc0, false, false);
        acc1 = __builtin_amdgcn_wmma_f32_16x16x4_f32(
            false, a, false, b1, (short)0, acc1, false, false);
      }
    }
  }

  float sc0 = nf[((size_t)b * OD + i) * OD + ja];
  float sc1 = nf[((size_t)b * OD + i) * OD + jb];
  int jr0 = j0 + n;
  int jr1 = j0 + 16 + n;
  if (jr0 < OD) {
    #pragma unroll
    for (int r = 0; r < 8; ++r) {
      int M = r + hi * 8;   // D VGPR r: lanes 0-15 -> row r, lanes 16-31 -> row r+8
      out[(((size_t)b * 16 + M) * OD + i) * OD + jr0] = acc0[r] * sc0;
    }
  }
  if (jr1 < OD) {
    #pragma unroll
    for (int r = 0; r < 8; ++r) {
      int M = r + hi * 8;
      out[(((size_t)b * 16 + M) * OD + i) * OD + jr1] = acc1[r] * sc1;
    }
  }
}

// ---------------------------------------------------------------------------
extern "C" void kernel_launch(void* const* d_in, const int* in_sizes, int n_in,
                              void* d_out, int out_size, void* d_ws, size_t ws_size,
                              hipStream_t stream) {
  (void)in_sizes; (void)n_in; (void)out_size; (void)ws_size;
  const float* x  = (const float*)d_in[0];
  const float* z  = (const float*)d_in[1];
  const float* fc = (const float*)d_in[2];
  float* out = (float*)d_out;

  float*  ws_f   = (float*)d_ws;
  float*  WA     = ws_f + WS_WA;
  float*  norm_d = ws_f + WS_NORM;
  float*  nf     = ws_f + WS_NF;
  double* S      = (double*)((char*)d_ws + WS_S_BYTE);

  // 1) filter synthesis (+ zero the reduction accumulators)
  wgt_kernel<<<dim3(256), dim3(64), 0, stream>>>(z, fc, WA, S);
  // 2) per-position sum of squares + global moments
  norm_kernel<<<dim3(OD, BB), dim3(256), 0, stream>>>(x, norm_d, S);
  // 3) normalization factor
  scale_kernel<<<dim3((ND + 255) / 256), dim3(256), 0, stream>>>(norm_d, S, nf);
  // 4) WMMA grouped conv (16x32 tile per wave), pre-scaled store
  conv_kernel<<<dim3(1, OD, BB), dim3(128), 0, stream>>>(x, WA, nf, out);
}